// LSTM_30296699306610
// MI455X (gfx1250) — compile-verified
//
#include <hip/hip_runtime.h>
#include <math.h>

// LSTM: T=128, B=32, IN=8000, H=35, L=4.  PyTorch gate order i,f,g,o.
// Input order (setup_inputs dict, lists flattened):
//   d_in[0]=x, d_in[1..4]=Wih[0..3], d_in[5..8]=Whh[0..3],
//   d_in[9..12]=bih[0..3], d_in[13..16]=bhh[0..3], d_in[17]=fcW, d_in[18]=fcb

typedef __attribute__((ext_vector_type(2))) float v2f;
typedef __attribute__((ext_vector_type(4))) float v4f;
typedef __attribute__((ext_vector_type(8))) float v8f;
typedef __attribute__((ext_vector_type(4))) int   v4i;

#define TSTEPS 128
#define BATCH  32
#define HID    35
#define G4H    140            // 4*H
#define NTILES 9              // ceil(144/16)
#define MROWS  4096           // T*B
#define XG_LD  144            // padded 4H stride
#define H_LD   36             // padded H stride
#define SCAN_THREADS 288      // 9 waves

#if __has_builtin(__builtin_amdgcn_global_load_async_to_lds_b128) && \
    __has_builtin(__builtin_amdgcn_s_wait_asynccnt)
#define HAVE_ASYNC_LDS 1
#define ASYNC_WAIT(n) __builtin_amdgcn_s_wait_asynccnt(n)
typedef __attribute__((address_space(1))) v4i* gptr_b128;   // global (prints as __device__)
typedef __attribute__((address_space(3))) v4i* lptr_b128;   // LDS    (prints as __shared__)
#else
#define HAVE_ASYNC_LDS 0
#define ASYNC_WAIT(n) ((void)0)
#endif

__device__ inline v8f wmma_f32(v2f a, v2f b, v8f c) {
    // D = A(16x4 f32) * B(4x16 f32) + C(16x16 f32)
    return __builtin_amdgcn_wmma_f32_16x16x4_f32(
        /*neg_a=*/false, a, /*neg_b=*/false, b,
        /*c_mod=*/(short)0, c, /*reuse_a=*/false, /*reuse_b=*/false);
}

__device__ inline float fast_sigmoid(float x) { return 1.f / (1.f + __expf(-x)); }
__device__ inline float fast_tanh(float x) {
    float e = __expf(-2.f * x);
    return (1.f - e) / (1.f + e);
}

// Copy one 18432-byte xg time-slice ([32][144] f32) into LDS.
// Async path: 4 x GLOBAL_LOAD_ASYNC_TO_LDS_B128 per thread (ASYNCcnt-tracked),
// lane-consecutive 16B chunks (coalesced).  Fallback: synchronous v4f copy.
__device__ inline void stage_xg_tile(const float* __restrict__ src,
                                     float* dst_lds, int tid) {
#pragma unroll
    for (int i = 0; i < 4; ++i) {
        const int c = tid + SCAN_THREADS * i;   // 0..1151 chunks of 16B
#if HAVE_ASYNC_LDS
        __builtin_amdgcn_global_load_async_to_lds_b128(
            (gptr_b128)((char*)src + (size_t)c * 16),
            (lptr_b128)((char*)dst_lds + (size_t)c * 16),
            /*offset=*/0, /*cpol=*/0);
#else
        ((v4f*)dst_lds)[c] = ((const v4f*)src)[c];
#endif
    }
}

// ---------------------------------------------------------------------------
// xg = A[4096,LDA(=KT)] @ W^T  + bih + bhh     (W is [140, KW], row-major)
// Output xg is [4096, 144] (cols 140..143 are exactly 0: zero-padded B frags,
// zero bias).  One wave = one 16-row M tile, all 9 N tiles (72 acc VGPRs).
// KT: padded K (multiple of 4), KW: true K of W rows, LDA: row stride of A.
// ---------------------------------------------------------------------------
template <int KT, int KW, int LDA>
__global__ void lstm_gemm_xg(const float* __restrict__ A,
                             const float* __restrict__ W,
                             const float* __restrict__ bih,
                             const float* __restrict__ bhh,
                             float* __restrict__ xg) {
    const int lane   = threadIdx.x & 31;
    const int wave   = threadIdx.x >> 5;          // 0..1 (block = 2 waves)
    const int lanelo = lane & 15;
    const int khalf  = lane >> 4;                 // 0: K=0,1  1: K=2,3
    const int mbase  = blockIdx.x * 32 + wave * 16;
    const int arow   = mbase + lanelo;

    constexpr bool KEXACT = (KW == KT);

    v8f acc[NTILES] = {};                          // zero-init accumulators

    for (int k = 0; k < KT; k += 4) {
        const int k0 = k + 2 * khalf;
        v2f a;
        a.x = A[arow * LDA + k0];
        a.y = A[arow * LDA + k0 + 1];
        if (KT > 64) {                             // big-K path: prefetch ahead
            __builtin_prefetch(&A[arow * LDA + ((k0 + 32 < KT) ? k0 + 32 : k0)], 0, 0);
        }
#pragma unroll
        for (int nt = 0; nt < NTILES; ++nt) {
            const int n = nt * 16 + lanelo;
            v2f b;
            if (KEXACT && nt < 8) {                // fully in-bounds (folds at compile time)
                b.x = W[n * KW + k0];
                b.y = W[n * KW + k0 + 1];
            } else {
                b.x = (n < G4H && (k0     < KW)) ? W[n * KW + k0]     : 0.f;
                b.y = (n < G4H && (k0 + 1 < KW)) ? W[n * KW + k0 + 1] : 0.f;
            }
            acc[nt] = wmma_f32(a, b, acc[nt]);
        }
    }

#pragma unroll
    for (int nt = 0; nt < NTILES; ++nt) {
        const int n    = nt * 16 + lanelo;
        const float bv = (n < G4H) ? (bih[n] + bhh[n]) : 0.f;
#pragma unroll
        for (int j = 0; j < 8; ++j) {
            const int row = mbase + j + 8 * khalf;
            xg[row * XG_LD + n] = acc[nt][j] + bv;
        }
    }
}

// ---------------------------------------------------------------------------
// Sequential LSTM scan for one layer.  Single workgroup, 9 waves (288 thr).
// Wave w owns N tile w for both M tiles (batch rows 0-15 / 16-31).
// h, c live in LDS; per step: g = xg[t] + h @ Whh^T via WMMA, then gates.
// xg time-slices are double-buffered in LDS and prefetched two steps ahead
// with GLOBAL_LOAD_ASYNC_TO_LDS_B128 so no global-load latency sits on the
// sequential critical path.
// hout is [T*B, 36] with pad col 35 zeroed (feeds next layer's GEMM, K=36).
// ---------------------------------------------------------------------------
__global__ void lstm_scan(const float* __restrict__ xg,    // [T*B, 144]
                          const float* __restrict__ Whh,   // [140, 35]
                          float* __restrict__ hout) {      // [T*B, 36]
    __shared__ float h_lds[BATCH][H_LD];
    __shared__ float c_lds[BATCH][H_LD];
    __shared__ float g_lds[BATCH][XG_LD];
    __shared__ float xg_lds[2][BATCH][XG_LD];     // double-buffered xg[t] tile

    const int tid    = threadIdx.x;
    const int lane   = tid & 31;
    const int wave   = tid >> 5;                  // 0..8 == N tile
    const int lanelo = lane & 15;
    const int khalf  = lane >> 4;
    const int n      = wave * 16 + lanelo;

    // Kick off async staging of xg[0] and xg[1] immediately.
    stage_xg_tile(xg,                            &xg_lds[0][0][0], tid);
    stage_xg_tile(xg + (size_t)BATCH * XG_LD,    &xg_lds[1][0][0], tid);

    // Preload Whh^T B-fragments for all 9 K-steps (zero-padded edges).
    v2f wb[NTILES];
#pragma unroll
    for (int ks = 0; ks < NTILES; ++ks) {
        const int k0 = ks * 4 + 2 * khalf;
        wb[ks].x = (n < G4H && (k0     < HID)) ? Whh[n * HID + k0]     : 0.f;
        wb[ks].y = (n < G4H && (k0 + 1 < HID)) ? Whh[n * HID + k0 + 1] : 0.f;
    }

    // h0 = c0 = 0 (pad col 35 stays 0 forever -> no K guard in WMMA A frags).
    for (int i = tid; i < BATCH * H_LD; i += SCAN_THREADS) {
        h_lds[i / H_LD][i % H_LD] = 0.f;
        c_lds[i / H_LD][i % H_LD] = 0.f;
    }
    ASYNC_WAIT(4);          // batch for t=0 landed (batch t=1 may be in flight)
    __syncthreads();

    for (int t = 0; t < TSTEPS; ++t) {
        const float (*xg_cur)[XG_LD] = xg_lds[t & 1];

        // Phase 1: g = xg[t] + h @ Whh^T  (uniform control flow, EXEC all-1s)
#pragma unroll
        for (int mt = 0; mt < 2; ++mt) {
            const int rbase = mt * 16;
            v8f acc;
#pragma unroll
            for (int j = 0; j < 8; ++j)
                acc[j] = xg_cur[rbase + j + 8 * khalf][n];
#pragma unroll
            for (int ks = 0; ks < NTILES; ++ks) {
                const int k0 = ks * 4 + 2 * khalf;
                v2f a;
                a.x = h_lds[rbase + lanelo][k0];
                a.y = h_lds[rbase + lanelo][k0 + 1];
                acc = wmma_f32(a, wb[ks], acc);
            }
#pragma unroll
            for (int j = 0; j < 8; ++j)
                g_lds[rbase + j + 8 * khalf][n] = acc[j];
        }
        __syncthreads();    // barrier A: phase-1 reads of xg_lds[t&1] complete

        // Phase 2: gates + state update.  32*36 = 1152 = 288 * 4 (uniform trip
        // count; j==35 lanes just zero the pad column of hout).
        for (int idx = tid; idx < BATCH * H_LD; idx += SCAN_THREADS) {
            const int b = idx / H_LD;
            const int j = idx % H_LD;
            if (j < HID) {
                const float i_ = fast_sigmoid(g_lds[b][j]);
                const float f_ = fast_sigmoid(g_lds[b][HID + j]);
                const float g_ = fast_tanh(g_lds[b][2 * HID + j]);
                const float o_ = fast_sigmoid(g_lds[b][3 * HID + j]);
                const float c  = f_ * c_lds[b][j] + i_ * g_;
                c_lds[b][j] = c;
                const float h = o_ * fast_tanh(c);
                h_lds[b][j] = h;
                hout[(size_t)(t * BATCH + b) * H_LD + j] = h;
            } else {
                hout[(size_t)(t * BATCH + b) * H_LD + j] = 0.f;
            }
        }

        // Prefetch xg[t+2] into the buffer just consumed (all waves passed
        // barrier A, so xg_lds[t&1] is dead).  Then make batch t+1 visible.
        if (t + 2 < TSTEPS) {
            stage_xg_tile(xg + (size_t)((t + 2) * BATCH) * XG_LD,
                          &xg_lds[t & 1][0][0], tid);
            ASYNC_WAIT(4);  // 8 in flight (t+1, t+2); <=4 => batch t+1 done
        } else {
            ASYNC_WAIT(0);  // drain: batch t+1 (if any) done
        }
        __syncthreads();    // barrier B: xg_lds[(t+1)&1] visible to all waves
    }
}

// ---------------------------------------------------------------------------
// y = h @ fcW^T + fcb ; out = log_softmax(y, axis=-1).   One thread per row.
// ---------------------------------------------------------------------------
__global__ void lstm_fc_logsoftmax(const float* __restrict__ h,    // [4096, 36]
                                   const float* __restrict__ fcW,  // [35, 35]
                                   const float* __restrict__ fcb,  // [35]
                                   float* __restrict__ out) {      // [4096, 35]
    __shared__ float Ws[HID * HID];
    __shared__ float bs[HID];
    for (int i = threadIdx.x; i < HID * HID; i += blockDim.x) Ws[i] = fcW[i];
    for (int i = threadIdx.x; i < HID; i += blockDim.x)       bs[i] = fcb[i];
    __syncthreads();

    const int r = blockIdx.x * blockDim.x + threadIdx.x;   // 0..4095

    float hr[HID];
#pragma unroll
    for (int k = 0; k < HID; ++k) hr[k] = h[(size_t)r * H_LD + k];

    float y[HID];
    float m = -3.0e38f;
#pragma unroll
    for (int j = 0; j < HID; ++j) {
        float s = bs[j];
#pragma unroll
        for (int k = 0; k < HID; ++k) s = fmaf(Ws[j * HID + k], hr[k], s);
        y[j] = s;
        m = fmaxf(m, s);
    }
    float sum = 0.f;
#pragma unroll
    for (int j = 0; j < HID; ++j) sum += __expf(y[j] - m);
    const float ls = m + __logf(sum);
#pragma unroll
    for (int j = 0; j < HID; ++j) out[(size_t)r * HID + j] = y[j] - ls;
}

// ---------------------------------------------------------------------------
extern "C" void kernel_launch(void* const* d_in, const int* in_sizes, int n_in,
                              void* d_out, int out_size, void* d_ws, size_t ws_size,
                              hipStream_t stream) {
    const float* x   = (const float*)d_in[0];
    const float* Wih[4] = {(const float*)d_in[1],  (const float*)d_in[2],
                           (const float*)d_in[3],  (const float*)d_in[4]};
    const float* Whh[4] = {(const float*)d_in[5],  (const float*)d_in[6],
                           (const float*)d_in[7],  (const float*)d_in[8]};
    const float* bih[4] = {(const float*)d_in[9],  (const float*)d_in[10],
                           (const float*)d_in[11], (const float*)d_in[12]};
    const float* bhh[4] = {(const float*)d_in[13], (const float*)d_in[14],
                           (const float*)d_in[15], (const float*)d_in[16]};
    const float* fcW = (const float*)d_in[17];
    const float* fcb = (const float*)d_in[18];
    float* out = (float*)d_out;

    // Workspace: xg [4096,144] f32 then h [4096,36] f32  (~2.95 MB total).
    float* xg   = (float*)d_ws;
    float* hbuf = xg + (size_t)MROWS * XG_LD;

    // Layer 0: big input projection (K = 8000), then scan.
    lstm_gemm_xg<8000, 8000, 8000><<<128, 64, 0, stream>>>(x, Wih[0], bih[0], bhh[0], xg);
    lstm_scan<<<1, SCAN_THREADS, 0, stream>>>(xg, Whh[0], hbuf);

    // Layers 1..3: small projection (K = 36 padded, h pad col is 0), then scan.
    for (int l = 1; l < 4; ++l) {
        lstm_gemm_xg<36, 35, 36><<<128, 64, 0, stream>>>(hbuf, Wih[l], bih[l], bhh[l], xg);
        lstm_scan<<<1, SCAN_THREADS, 0, stream>>>(xg, Whh[l], hbuf);
    }

    // FC + log_softmax.
    lstm_fc_logsoftmax<<<64, 64, 0, stream>>>(hbuf, fcW, fcb, out);
}